// HeteroGNN_11974368821561
// MI455X (gfx1250) — compile-verified
//
#include <hip/hip_runtime.h>
#include <hip/hip_bf16.h>

typedef __attribute__((ext_vector_type(2))) float v2f;
typedef __attribute__((ext_vector_type(8))) float v8f;

#define N_NODES 50000
#define N_EDGES 800000
#define HIDDEN  64
#define OUTD    16

// ---------------- degree ----------------
__global__ void deg_count_kernel(const long long* __restrict__ ei,
                                 float* __restrict__ deg) {
    int i = blockIdx.x * blockDim.x + threadIdx.x;
    int stride = gridDim.x * blockDim.x;
    for (int e = i; e < N_EDGES; e += stride) {
        long long d = ei[N_EDGES + e];
        atomicAdd(&deg[d], 1.0f);
    }
}

__global__ void deg_inv_kernel(float* __restrict__ deg) {
    int i = blockIdx.x * blockDim.x + threadIdx.x;
    if (i < N_NODES) deg[i] = 1.0f / fmaxf(deg[i], 1.0f);
}

// ---------------- scatter-sum: one wave32 per edge ----------------
__global__ __launch_bounds__(256)
void scatter_kernel(const long long* __restrict__ ei,
                    const float* __restrict__ x,
                    float* __restrict__ agg) {
    int gid   = blockIdx.x * blockDim.x + threadIdx.x;
    int lane  = gid & 31;
    int wave  = gid >> 5;
    int nwave = (gridDim.x * blockDim.x) >> 5;
    for (int e = wave; e < N_EDGES; e += nwave) {
        long long s = ei[e];
        long long d = ei[N_EDGES + e];
        // one coalesced 256B row read per edge (32 lanes x float2)
        const float2* xs = (const float2*)(x + s * HIDDEN);
        float2 v = xs[lane];
        float* ad = agg + d * HIDDEN + lane * 2;
        atomicAdd(ad + 0, v.x);
        atomicAdd(ad + 1, v.y);
    }
}

// ---------------- fused SAGE layer GEMM: x' = relu(agg*dinv @ Wl + bl + x @ Wr)
// 256 threads = 8 waves; each wave computes a 16-row x 64-col strip via
// V_WMMA_F32_16X16X4_F32 (4 N-tiles, K=64 in 16 steps, two K-passes).
// Weights are staged in LDS pre-packed in WMMA-B operand order so each B
// operand is a single aligned ds_load_b64:
//   sW[(k>>1)*128 + n*2 + (k&1)] = W[k*64 + n]
__global__ __launch_bounds__(256)
void sage_gemm_kernel(const float* __restrict__ agg,
                      const float* __restrict__ dinv,
                      const float* __restrict__ xin,
                      const float* __restrict__ Wl,
                      const float* __restrict__ bl,
                      const float* __restrict__ Wr,
                      float* __restrict__ xout) {
    __shared__ __align__(16) float sWl[HIDDEN * HIDDEN];
    __shared__ __align__(16) float sWr[HIDDEN * HIDDEN];
    __shared__ float sB[HIDDEN];

    int tid = threadIdx.x;
    for (int i = tid; i < HIDDEN * HIDDEN; i += 256) {
        int k = i >> 6;
        int n = i & 63;
        int dst = ((k >> 1) << 7) + (n << 1) + (k & 1);  // paired-K layout
        sWl[dst] = Wl[i];
        sWr[dst] = Wr[i];
    }
    if (tid < HIDDEN) sB[tid] = bl[tid];
    __syncthreads();

    int wave   = tid >> 5;
    int lane   = tid & 31;
    int lane16 = lane & 15;
    int hi     = lane >> 4;           // 0: K+{0,1}, 1: K+{2,3}
    int m0     = blockIdx.x * 128 + wave * 16;

    int mrow = m0 + lane16;
    if (mrow >= N_NODES) mrow = N_NODES - 1;   // clamp: keep EXEC all-ones for WMMA
    float di = dinv[mrow];
    const float* arow = agg + (long long)mrow * HIDDEN;
    const float* xrow = xin + (long long)mrow * HIDDEN;

    v8f c[4];
    c[0] = (v8f){}; c[1] = (v8f){}; c[2] = (v8f){}; c[3] = (v8f){};

    // pass 1: (agg * dinv) @ Wl
    #pragma unroll
    for (int kb = 0; kb < HIDDEN; kb += 4) {
        int k0 = kb + 2 * hi;
        int kp = (kb >> 1) + hi;      // paired-K row in LDS
        v2f a;
        a.x = arow[k0]     * di;
        a.y = arow[k0 + 1] * di;
        #pragma unroll
        for (int t = 0; t < 4; ++t) {
            int n = t * 16 + lane16;
            v2f b = *(const v2f*)&sWl[(kp << 7) + (n << 1)];
            c[t] = __builtin_amdgcn_wmma_f32_16x16x4_f32(
                false, a, false, b, (short)0, c[t], false, false);
        }
    }
    // pass 2: x @ Wr
    #pragma unroll
    for (int kb = 0; kb < HIDDEN; kb += 4) {
        int k0 = kb + 2 * hi;
        int kp = (kb >> 1) + hi;
        v2f a;
        a.x = xrow[k0];
        a.y = xrow[k0 + 1];
        #pragma unroll
        for (int t = 0; t < 4; ++t) {
            int n = t * 16 + lane16;
            v2f b = *(const v2f*)&sWr[(kp << 7) + (n << 1)];
            c[t] = __builtin_amdgcn_wmma_f32_16x16x4_f32(
                false, a, false, b, (short)0, c[t], false, false);
        }
    }

    // epilogue: bias + relu + store (C layout: VGPR j -> row hi*8+j, col lane16)
    #pragma unroll
    for (int t = 0; t < 4; ++t) {
        int n = t * 16 + lane16;
        float bias = sB[n];
        #pragma unroll
        for (int j = 0; j < 8; ++j) {
            int row = m0 + hi * 8 + j;
            if (row < N_NODES) {
                float v = c[t][j] + bias;
                xout[(long long)row * HIDDEN + n] = fmaxf(v, 0.0f);
            }
        }
    }
}

// ---------------- output head: y = x @ W_out + b_out  (64x16) ----------------
__global__ __launch_bounds__(256)
void out_gemm_kernel(const float* __restrict__ xin,
                     const float* __restrict__ Wo,
                     const float* __restrict__ bo,
                     float* __restrict__ out) {
    __shared__ __align__(16) float sW[HIDDEN * OUTD];
    __shared__ float sB[OUTD];

    int tid = threadIdx.x;
    for (int i = tid; i < HIDDEN * OUTD; i += 256) {
        int k = i >> 4;
        int n = i & 15;
        int dst = ((k >> 1) << 5) + (n << 1) + (k & 1);  // paired-K layout
        sW[dst] = Wo[i];
    }
    if (tid < OUTD) sB[tid] = bo[tid];
    __syncthreads();

    int wave   = tid >> 5;
    int lane   = tid & 31;
    int lane16 = lane & 15;
    int hi     = lane >> 4;
    int m0     = blockIdx.x * 128 + wave * 16;

    int mrow = m0 + lane16;
    if (mrow >= N_NODES) mrow = N_NODES - 1;
    const float* xrow = xin + (long long)mrow * HIDDEN;

    v8f c = (v8f){};
    #pragma unroll
    for (int kb = 0; kb < HIDDEN; kb += 4) {
        int k0 = kb + 2 * hi;
        int kp = (kb >> 1) + hi;
        v2f a;
        a.x = xrow[k0];
        a.y = xrow[k0 + 1];
        v2f b = *(const v2f*)&sW[(kp << 5) + (lane16 << 1)];
        c = __builtin_amdgcn_wmma_f32_16x16x4_f32(
            false, a, false, b, (short)0, c, false, false);
    }

    float bias = sB[lane16];
    #pragma unroll
    for (int j = 0; j < 8; ++j) {
        int row = m0 + hi * 8 + j;
        if (row < N_NODES) {
            out[(long long)row * OUTD + lane16] = c[j] + bias;
        }
    }
}

extern "C" void kernel_launch(void* const* d_in, const int* in_sizes, int n_in,
                              void* d_out, int out_size, void* d_ws, size_t ws_size,
                              hipStream_t stream) {
    const float*     x    = (const float*)d_in[0];
    const long long* ei   = (const long long*)d_in[1];   // int64 edge_index [2, E]
    const float*     Wl   = (const float*)d_in[2];       // [3, 64, 64]
    const float*     bl   = (const float*)d_in[3];       // [3, 64]
    const float*     Wr   = (const float*)d_in[4];       // [3, 64, 64]
    const float*     Wo   = (const float*)d_in[5];       // [64, 16]
    const float*     bo   = (const float*)d_in[6];       // [16]
    float*           out  = (float*)d_out;

    float* ws = (float*)d_ws;
    size_t nh = (size_t)N_NODES * HIDDEN;
    float* deg = ws;                            // N_NODES (becomes 1/deg)
    float* agg = ws + ((N_NODES + 63) & ~63);   // N*H
    float* xb0 = agg + nh;                      // N*H
    float* xb1 = xb0 + nh;                      // N*H

    // degree (recomputed each call: deterministic)
    hipMemsetAsync(deg, 0, N_NODES * sizeof(float), stream);
    deg_count_kernel<<<1024, 256, 0, stream>>>(ei, deg);
    deg_inv_kernel<<<(N_NODES + 255) / 256, 256, 0, stream>>>(deg);

    const float* xin = x;
    float* bufs[2] = {xb0, xb1};
    for (int l = 0; l < 3; ++l) {
        hipMemsetAsync(agg, 0, nh * sizeof(float), stream);
        scatter_kernel<<<2048, 256, 0, stream>>>(ei, xin, agg);
        float* xo = bufs[l & 1];
        sage_gemm_kernel<<<(N_NODES + 127) / 128, 256, 0, stream>>>(
            agg, deg, xin,
            Wl + (size_t)l * HIDDEN * HIDDEN,
            bl + (size_t)l * HIDDEN,
            Wr + (size_t)l * HIDDEN * HIDDEN,
            xo);
        xin = xo;
    }

    out_gemm_kernel<<<(N_NODES + 127) / 128, 256, 0, stream>>>(xin, Wo, bo, out);
}